// FractalDepthQuaternionWuBuRungs_76888504533225
// MI455X (gfx1250) — compile-verified
//
#include <hip/hip_runtime.h>
#include <math.h>

typedef __attribute__((ext_vector_type(2))) float v2f;
typedef __attribute__((ext_vector_type(4))) float v4f;
typedef __attribute__((ext_vector_type(8))) float v8f;

#define DIN 512
#define DOUT 512
#define NRUNG 16
#define TCLAMP 10000.0f
#define QEPS 1e-5f

__device__ __forceinline__ float clampT(float v) {
    return fminf(fmaxf(v, -TCLAMP), TCLAMP);
}

// Precompute the 16 (p,q) unit quaternions from so3 params into ws.
__global__ void quat_setup_kernel(const float* __restrict__ so3_p,
                                  const float* __restrict__ so3_q,
                                  float* __restrict__ pq) {
    int i = threadIdx.x;            // 0..31 -> (rung, p/q)
    if (i < 2 * NRUNG) {
        int r = i >> 1, which = i & 1;
        const float* v = (which ? so3_q : so3_p) + r * 3;
        float vx = v[0], vy = v[1], vz = v[2];
        float ang = sqrtf(vx * vx + vy * vy + vz * vz);
        float w = 1.0f, x = 0.0f, y = 0.0f, z = 0.0f;
        if (ang >= QEPS) {
            float inv  = 1.0f / fmaxf(ang, QEPS);
            float half = 0.5f * ang;
            float s    = sinf(half) * inv;
            w = cosf(half); x = vx * s; y = vy * s; z = vz * s;
        }
        float* o = pq + (r * 2 + which) * 4;
        o[0] = w; o[1] = x; o[2] = y; o[3] = z;
    }
}

__launch_bounds__(256, 2)
__global__ void wubu_kernel(const float* __restrict__ x,
                            const float* __restrict__ in_W,
                            const float* __restrict__ in_b,
                            const float* __restrict__ pq,
                            const float* __restrict__ W1,
                            const float* __restrict__ b1,
                            const float* __restrict__ W2,
                            const float* __restrict__ b2,
                            const float* __restrict__ out_W,
                            const float* __restrict__ out_b,
                            float* __restrict__ out) {
    const int lane = threadIdx.x & 31;
    const int wave = threadIdx.x >> 5;
    const long tok_base = (long)blockIdx.x * 256 + (long)wave * 32;

    // ---- Phase 1: h = clip(x @ in_W + in_b) for this wave's 32 tokens ----
    // Lane's k-slice: k in {4*lane + 128*j + i, i<4, j<4} (16 rows of in_W,
    // held in 64 VGPRs, reused across all 32 tokens).
    const v4f* inW4 = (const v4f*)in_W;         // row k -> 4 columns
    v4f w[4][4];
#pragma unroll
    for (int j = 0; j < 4; j++)
#pragma unroll
        for (int i = 0; i < 4; i++)
            w[j][i] = inW4[4 * lane + 128 * j + i];

    const float ib0 = in_b[0], ib1 = in_b[1], ib2 = in_b[2], ib3 = in_b[3];

    float h0 = 0.f, h1 = 0.f, h2 = 0.f, h3 = 0.f;   // lane t owns token t

    // Double-buffered, non-temporal, fully coalesced row loads.
    const v4f* xr = (const v4f*)(x + tok_base * DIN);
    v4f nxt0 = __builtin_nontemporal_load(xr + lane);
    v4f nxt1 = __builtin_nontemporal_load(xr + lane + 32);
    v4f nxt2 = __builtin_nontemporal_load(xr + lane + 64);
    v4f nxt3 = __builtin_nontemporal_load(xr + lane + 96);

    for (int t = 0; t < 32; t++) {
        v4f xq0 = nxt0, xq1 = nxt1, xq2 = nxt2, xq3 = nxt3;
        const int tn = (t < 31) ? (t + 1) : 31;     // issue next row's loads now
        const v4f* xn = (const v4f*)(x + (tok_base + tn) * DIN);
        nxt0 = __builtin_nontemporal_load(xn + lane);
        nxt1 = __builtin_nontemporal_load(xn + lane + 32);
        nxt2 = __builtin_nontemporal_load(xn + lane + 64);
        nxt3 = __builtin_nontemporal_load(xn + lane + 96);

        float a0 = 0.f, a1 = 0.f, a2 = 0.f, a3 = 0.f;
#define ACC4(XQ, J)                                                         \
        a0 = fmaf(XQ.x, w[J][0].x, a0); a1 = fmaf(XQ.x, w[J][0].y, a1);     \
        a2 = fmaf(XQ.x, w[J][0].z, a2); a3 = fmaf(XQ.x, w[J][0].w, a3);     \
        a0 = fmaf(XQ.y, w[J][1].x, a0); a1 = fmaf(XQ.y, w[J][1].y, a1);     \
        a2 = fmaf(XQ.y, w[J][1].z, a2); a3 = fmaf(XQ.y, w[J][1].w, a3);     \
        a0 = fmaf(XQ.z, w[J][2].x, a0); a1 = fmaf(XQ.z, w[J][2].y, a1);     \
        a2 = fmaf(XQ.z, w[J][2].z, a2); a3 = fmaf(XQ.z, w[J][2].w, a3);     \
        a0 = fmaf(XQ.w, w[J][3].x, a0); a1 = fmaf(XQ.w, w[J][3].y, a1);     \
        a2 = fmaf(XQ.w, w[J][3].z, a2); a3 = fmaf(XQ.w, w[J][3].w, a3);
        ACC4(xq0, 0) ACC4(xq1, 1) ACC4(xq2, 2) ACC4(xq3, 3)
#undef ACC4

#pragma unroll
        for (int off = 16; off >= 1; off >>= 1) {
            a0 += __shfl_xor(a0, off, 32);
            a1 += __shfl_xor(a1, off, 32);
            a2 += __shfl_xor(a2, off, 32);
            a3 += __shfl_xor(a3, off, 32);
        }
        h0 = (lane == t) ? clampT(a0 + ib0) : h0;
        h1 = (lane == t) ? clampT(a1 + ib1) : h1;
        h2 = (lane == t) ? clampT(a2 + ib2) : h2;
        h3 = (lane == t) ? clampT(a3 + ib3) : h3;
    }

    // ---- Phase 2: 16 rungs (uniform params -> SGPR scalar loads) ----
    for (int r = 0; r < NRUNG; r++) {
        const float* P = pq + r * 8;
        const float pw = P[0], px = P[1], py = P[2], pz = P[3];
        const float qw = P[4], qx = P[5], qy = P[6], qz = P[7];
        // t = p * h   (Hamilton)
        float tw = pw * h0 - px * h1 - py * h2 - pz * h3;
        float tx = pw * h1 + px * h0 + py * h3 - pz * h2;
        float ty = pw * h2 - px * h3 + py * h0 + pz * h1;
        float tz = pw * h3 + px * h2 - py * h1 + pz * h0;
        // hr = t * q
        float rw = tw * qw - tx * qx - ty * qy - tz * qz;
        float rx = tw * qx + tx * qw + ty * qz - tz * qy;
        float ry = tw * qy - tx * qz + ty * qw + tz * qx;
        float rz = tw * qz + tx * qy - ty * qx + tz * qw;

        const float* w1  = W1 + r * 16;
        const float* bb1 = b1 + r * 4;
        const float* w2  = W2 + r * 16;
        const float* bb2 = b2 + r * 4;
        float m[4], g[4];
#pragma unroll
        for (int i = 0; i < 4; i++)
            m[i] = bb1[i] + rw * w1[0 * 4 + i] + rx * w1[1 * 4 + i]
                          + ry * w1[2 * 4 + i] + rz * w1[3 * 4 + i];
#pragma unroll
        for (int i = 0; i < 4; i++)   // exact GELU
            g[i] = 0.5f * m[i] * (1.0f + erff(m[i] * 0.70710678118654752f));
        h0 = clampT(bb2[0] + g[0] * w2[0] + g[1] * w2[4] + g[2] * w2[8]  + g[3] * w2[12]);
        h1 = clampT(bb2[1] + g[0] * w2[1] + g[1] * w2[5] + g[2] * w2[9]  + g[3] * w2[13]);
        h2 = clampT(bb2[2] + g[0] * w2[2] + g[1] * w2[6] + g[2] * w2[10] + g[3] * w2[14]);
        h3 = clampT(bb2[3] + g[0] * w2[3] + g[1] * w2[7] + g[2] * w2[11] + g[3] * w2[15]);
    }

    // ---- Phase 3: out = h @ out_W + out_b via V_WMMA_F32_16X16X4_F32 ----
    // A 16x4 f32 layout: lanes 0-15 hold M=lane with VGPR0=K0, VGPR1=K1;
    //                    lanes 16-31 hold M=lane-16 with VGPR0=K2, VGPR1=K3.
    const int  cl = lane & 15;
    const bool lo = lane < 16;
    float s00 = __shfl(h0, cl, 32),      s01 = __shfl(h1, cl, 32);
    float s02 = __shfl(h2, cl, 32),      s03 = __shfl(h3, cl, 32);
    float s10 = __shfl(h0, 16 + cl, 32), s11 = __shfl(h1, 16 + cl, 32);
    float s12 = __shfl(h2, 16 + cl, 32), s13 = __shfl(h3, 16 + cl, 32);
    v2f A0, A1;
    A0[0] = lo ? s00 : s02;  A0[1] = lo ? s01 : s03;   // tokens 0..15
    A1[0] = lo ? s10 : s12;  A1[1] = lo ? s11 : s13;   // tokens 16..31

    // B 4x16 f32 layout: VGPR0 lanes0-15=K0 row, lanes16-31=K2; VGPR1 K1/K3.
    const int kb = lo ? 0 : 2;
    // Single store base: both D tiles live within the 24-bit immediate window
    // (row v -> v*2048 B, second tile -> +32768 B).
    float* ob = out + (tok_base + (lo ? 0 : 8)) * DOUT;

    for (int ct = 0; ct < 32; ct++) {
        const int col = ct * 16 + cl;
        v2f Bv;
        Bv[0] = out_W[kb * DOUT + col];
        Bv[1] = out_W[(kb + 1) * DOUT + col];
        const float bias = out_b[col];      // C[m][n] = out_b[n]
        v8f C;
#pragma unroll
        for (int v = 0; v < 8; v++) C[v] = bias;

        v8f D0 = __builtin_amdgcn_wmma_f32_16x16x4_f32(
            false, A0, false, Bv, (short)0, C, false, false);
        v8f D1 = __builtin_amdgcn_wmma_f32_16x16x4_f32(
            false, A1, false, Bv, (short)0, C, false, false);

        // D layout: VGPR v -> row M = v + (lane<16 ? 0 : 8), col = lane&15
        float* op = ob + col;
#pragma unroll
        for (int v = 0; v < 8; v++)
            __builtin_nontemporal_store(D0[v], op + v * DOUT);
#pragma unroll
        for (int v = 0; v < 8; v++)
            __builtin_nontemporal_store(D1[v], op + (16 + v) * DOUT);
    }
}

extern "C" void kernel_launch(void* const* d_in, const int* in_sizes, int n_in,
                              void* d_out, int out_size, void* d_ws, size_t ws_size,
                              hipStream_t stream) {
    const float* x     = (const float*)d_in[0];
    const float* in_W  = (const float*)d_in[1];
    const float* in_b  = (const float*)d_in[2];
    const float* so3_p = (const float*)d_in[3];
    const float* so3_q = (const float*)d_in[4];
    const float* W1    = (const float*)d_in[5];
    const float* b1    = (const float*)d_in[6];
    const float* W2    = (const float*)d_in[7];
    const float* b2    = (const float*)d_in[8];
    const float* out_W = (const float*)d_in[9];
    const float* out_b = (const float*)d_in[10];
    float* out = (float*)d_out;
    float* pqws = (float*)d_ws;     // 128 floats: 16 rungs x (p,q) x 4

    quat_setup_kernel<<<1, 32, 0, stream>>>(so3_p, so3_q, pqws);

    // 131072 tokens / (8 waves * 32 tokens) = 512 blocks of 256 threads
    wubu_kernel<<<dim3(512), dim3(256), 0, stream>>>(
        x, in_W, in_b, pqws, W1, b1, W2, b2, out_W, out_b, out);
}